// PRMPHeteroGNN_1099511628114
// MI455X (gfx1250) — compile-verified
//
#include <hip/hip_runtime.h>
#include <hip/hip_bf16.h>

// ---------------------------------------------------------------------------
// PRMP hetero-GNN for MI455X (gfx1250, wave32).
//
// GEMMs run on V_WMMA_F32_16X16X4_F32 (fp32 in/out -> reference numerics).
// Layout: one wave owns one 16-col N-tile; its B operand (K=128 slice of W)
// lives in 64 VGPRs for the whole block. The 16x128 A tile is shared by all
// 8 waves in LDS. For contiguous A (node GEMMs) the tile is staged by the
// Tensor Data Mover (tensor_load_to_lds + s_wait_tensorcnt) with LDS double
// buffering so the DMA of tile t+1 overlaps the 256 WMMAs of tile t. For
// gathered A (edge MLP / head) coalesced float4 staging is used instead
// (TDM gather mode caps at 8 rows of 32-bit indices).
//
// Segment means: f32 global atomics (accumulators 51/205 MB -> L2-resident,
// 192 MB). LayerNorm/head: one wave per row, shfl_xor reductions (wave32).
// ---------------------------------------------------------------------------

typedef __attribute__((ext_vector_type(2))) float v2f;
typedef __attribute__((ext_vector_type(8))) float v8f;
typedef __attribute__((ext_vector_type(4))) unsigned int tdm_v4u;
typedef __attribute__((ext_vector_type(8))) int tdm_v8i;
typedef __attribute__((ext_vector_type(4))) int tdm_v4i;

#define GNN_H 128
#define STEPS 8   // row tiles processed per block

#if defined(__gfx1250__) && __has_builtin(__builtin_amdgcn_tensor_load_to_lds)
#define HAVE_TDM 1
#else
#define HAVE_TDM 0
#endif

#if HAVE_TDM
// Issue a 2D TDM load: `rows` x 128 f32 tile at gptr (row stride 128 f32)
// into LDS byte offset lds_off. tensor_dim1 = M so OOB rows read zero.
// 6-arg builtin form: (v4u g0, v8i g1, v4i g2, v4i g3, v8i g4, i32 cpol).
__device__ __forceinline__ void tdm_issue_2d(unsigned lds_off, const float* gptr,
                                             int rows, int M_rows) {
  unsigned long long ga = (unsigned long long)(uintptr_t)gptr;
  tdm_v4u g0;
  g0[0] = 1u;                                                   // count=1, user D#
  g0[1] = lds_off;                                              // lds_addr (bytes)
  g0[2] = (unsigned)(ga & 0xFFFFFFFFull);                       // global_addr lo
  g0[3] = (unsigned)((ga >> 32) & 0x1FFFFFFull) | (2u << 30);   // ga[56:32] | type=2
  tdm_v8i g1;
  g1[0] = (int)(2u << 16);                                      // data_size = 4B
  g1[1] = (int)((128u & 0xFFFFu) << 16);                        // tensor_dim0 lo16
  g1[2] = (int)((128u >> 16) | ((unsigned)M_rows << 16));       // td0 hi | td1 lo
  g1[3] = (int)((((unsigned)M_rows) >> 16) | (128u << 16));     // td1 hi | tile_dim0
  g1[4] = (int)((unsigned)rows);                                // tile_dim1 | tile_dim2=0
  g1[5] = (int)128u;                                            // tensor_dim0_stride lo32
  g1[6] = 0;                                                    // stride hi | td1_stride lo
  g1[7] = 0;                                                    // td1_stride hi
  tdm_v4i z4;
  z4[0] = 0; z4[1] = 0; z4[2] = 0; z4[3] = 0;
  tdm_v8i z8;
#pragma unroll
  for (int i = 0; i < 8; ++i) z8[i] = 0;
  __builtin_amdgcn_tensor_load_to_lds(g0, g1, z4, z4, z8, 0);
}
#endif

// ---------------------------------------------------------------------------
// GEMM: Y[M,N] = act( X[gidx[m],0:128] @ W[128,N] + bias (+Y if accum) )
// K = 128. Block = 256 threads = 8 waves. Wave w -> N-tile (w % NT),
// row-group (w / NT). Block covers STEPS * (GROUPS*16) rows.
// ---------------------------------------------------------------------------
template <int NCOLS, bool USE_TDM>
__global__ __launch_bounds__(256)
void gemm_wmma_bk(const float* __restrict__ X, const int* __restrict__ gidx,
                  const float* __restrict__ W, const float* __restrict__ bias,
                  float* __restrict__ Y, int M, int relu, int accum) {
  constexpr int NT = NCOLS / 16;       // N-tiles (8 or 4)
  constexpr int GROUPS = 8 / NT;       // row groups per step (1 or 2)
  constexpr int ROWS = GROUPS * 16;    // rows staged per step
  constexpr int NBUF = USE_TDM ? 2 : 1;
  __shared__ __align__(16) float sA[NBUF * ROWS * GNN_H];

  const int tid = threadIdx.x;
  const int wave = tid >> 5;
  const int lane = tid & 31;
  const int half = lane >> 4;
  const int m = lane & 15;
  const int nt = wave % NT;
  const int grp = wave / NT;
  const int col = nt * 16 + m;
  const int blockBase = blockIdx.x * (STEPS * ROWS);

  // --- B operand (4x16 f32 layout) for this wave's N-tile, all K, in VGPRs.
  // VGPR0: rows {k | k+2}, VGPR1: rows {k+1 | k+3} (lanes 0-15 | 16-31).
  v2f breg[32];
#pragma unroll
  for (int kk2 = 0; kk2 < 32; ++kk2) {
    int k0 = kk2 * 4 + 2 * half;
    breg[kk2].x = W[(size_t)k0 * NCOLS + col];
    breg[kk2].y = W[(size_t)(k0 + 1) * NCOLS + col];
  }
  const float bv = bias ? bias[col] : 0.0f;

  auto stage = [&](int step, int bufOff) {  // vector-load staging (gather path)
    const int rowbase = blockBase + step * ROWS;
    for (int f = tid; f < ROWS * 32; f += 256) {
      int r = f >> 5, cw = f & 31;
      int gr = rowbase + r;
      float4 v = make_float4(0.f, 0.f, 0.f, 0.f);
      if (gr < M) {
        int sr = gidx ? gidx[gr] : gr;
        v = *(const float4*)(X + (size_t)sr * GNN_H + cw * 4);
      }
      *(float4*)(&sA[bufOff + r * GNN_H + cw * 4]) = v;
    }
  };

  auto compute = [&](int step, int bufOff) {
    const int rowstart = blockBase + step * ROWS + grp * 16;
    const float* aBase = &sA[bufOff + grp * 16 * GNN_H + m * GNN_H + 2 * half];
    v8f c;
#pragma unroll
    for (int j = 0; j < 8; ++j) c[j] = 0.0f;
#pragma unroll
    for (int kk2 = 0; kk2 < 32; ++kk2) {
      v2f a = *(const v2f*)(aBase + kk2 * 4);  // A row m, k pair (8B aligned)
      c = __builtin_amdgcn_wmma_f32_16x16x4_f32(false, a, false, breg[kk2],
                                                (short)0, c, false, false);
    }
    // C layout: VGPR i -> row i (lanes 0-15) / row i+8 (lanes 16-31), col = lane
#pragma unroll
    for (int i = 0; i < 8; ++i) {
      int row = rowstart + i + half * 8;
      if (row < M) {
        size_t o = (size_t)row * NCOLS + col;
        float v = c[i] + bv;
        if (accum) v += Y[o];
        if (relu) v = fmaxf(v, 0.0f);
        Y[o] = v;
      }
    }
  };

  if constexpr (USE_TDM) {
#if HAVE_TDM
    // Double-buffered TDM pipeline: tile t+1 DMA overlaps tile t WMMAs.
    if (wave == 0) {
      int rows = M - blockBase;
      if (rows > 0) tdm_issue_2d((unsigned)(uintptr_t)&sA[0],
                                 X + (size_t)blockBase * GNN_H,
                                 rows < ROWS ? rows : ROWS, M);
      __builtin_amdgcn_s_wait_tensorcnt(0);
    }
    __syncthreads();
    for (int s = 0; s < STEPS; ++s) {
      if (wave == 0 && s + 1 < STEPS) {
        int nb = blockBase + (s + 1) * ROWS;
        int rows = M - nb;
        if (rows > 0) tdm_issue_2d((unsigned)(uintptr_t)&sA[((s + 1) & 1) * ROWS * GNN_H],
                                   X + (size_t)nb * GNN_H,
                                   rows < ROWS ? rows : ROWS, M);
      }
      if (blockBase + s * ROWS < M) compute(s, (s & 1) * ROWS * GNN_H);
      if (wave == 0) __builtin_amdgcn_s_wait_tensorcnt(0);
      __syncthreads();
    }
#endif
  } else {
    for (int s = 0; s < STEPS; ++s) {
      __syncthreads();
      if (blockBase + s * ROWS < M) {
        stage(s, 0);
        __syncthreads();
        compute(s, 0);
      } else {
        __syncthreads();
      }
    }
  }
}

// ---------------------------------------------------------------------------
// Elementwise / scatter / reduce kernels
// ---------------------------------------------------------------------------
__global__ void zero_f32(float* __restrict__ p, size_t n) {
  size_t i = (size_t)blockIdx.x * blockDim.x + threadIdx.x;
  if (i < n) p[i] = 0.0f;
}

__global__ void scatter_residual(const float* __restrict__ P, const float* __restrict__ HT,
                                 const int* __restrict__ src, const int* __restrict__ dst,
                                 float* __restrict__ seg, float* __restrict__ cnt, int E) {
  int t = blockIdx.x * blockDim.x + threadIdx.x;
  int e = t >> 5, lane = t & 31;
  if (e >= E) return;
  int s = src[e], d = dst[e];
  size_t pb = (size_t)e * GNN_H, hb = (size_t)d * GNN_H, sb = (size_t)s * GNN_H;
#pragma unroll
  for (int j = 0; j < 4; ++j) {
    int c = lane * 4 + j;
    atomicAdd(&seg[sb + c], HT[hb + c] - P[pb + c]);
  }
  if (lane == 0) atomicAdd(&cnt[s], 1.0f);
}

__global__ void scatter_gather(const float* __restrict__ tab,
                               const int* __restrict__ src, const int* __restrict__ dst,
                               float* __restrict__ seg, float* __restrict__ cnt, int E) {
  int t = blockIdx.x * blockDim.x + threadIdx.x;
  int e = t >> 5, lane = t & 31;
  if (e >= E) return;
  int s = src[e], d = dst[e];
  size_t tb = (size_t)s * GNN_H, sb = (size_t)d * GNN_H;
#pragma unroll
  for (int j = 0; j < 4; ++j) {
    int c = lane * 4 + j;
    atomicAdd(&seg[sb + c], tab[tb + c]);
  }
  if (lane == 0) atomicAdd(&cnt[d], 1.0f);
}

__global__ void seg_div(float* __restrict__ seg, const float* __restrict__ cnt, int n) {
  int t = blockIdx.x * blockDim.x + threadIdx.x;
  int node = t >> 5, lane = t & 31;
  if (node >= n) return;
  float inv = 1.0f / fmaxf(cnt[node], 1.0f);
  size_t b = (size_t)node * GNN_H;
#pragma unroll
  for (int j = 0; j < 4; ++j) seg[b + lane * 4 + j] *= inv;
}

// out = act( LN( (A [+ B]) * scale ) * g + b ); one wave per row
__global__ __launch_bounds__(256)
void ln_kernel(const float* __restrict__ A, const float* __restrict__ B, float scale,
               const float* __restrict__ g, const float* __restrict__ bb,
               float* __restrict__ out, int n, int relu) {
  int wave = threadIdx.x >> 5, lane = threadIdx.x & 31;
  int row = blockIdx.x * 8 + wave;
  if (row >= n) return;
  size_t base = (size_t)row * GNN_H;
  float x[4];
  float s = 0.0f;
#pragma unroll
  for (int j = 0; j < 4; ++j) {
    int c = lane * 4 + j;
    float v = A[base + c];
    if (B) v += B[base + c];
    v *= scale;
    x[j] = v;
    s += v;
  }
#pragma unroll
  for (int off = 16; off > 0; off >>= 1) s += __shfl_xor(s, off, 32);
  float mean = s * (1.0f / 128.0f);
  float vs = 0.0f;
#pragma unroll
  for (int j = 0; j < 4; ++j) {
    float d = x[j] - mean;
    vs += d * d;
  }
#pragma unroll
  for (int off = 16; off > 0; off >>= 1) vs += __shfl_xor(vs, off, 32);
  float inv = rsqrtf(vs * (1.0f / 128.0f) + 1e-5f);
#pragma unroll
  for (int j = 0; j < 4; ++j) {
    int c = lane * 4 + j;
    float y = (x[j] - mean) * inv * g[c] + bb[c];
    if (relu) y = fmaxf(y, 0.0f);
    out[base + c] = y;
  }
}

__global__ __launch_bounds__(256)
void head_dot(const float* __restrict__ T, const float* __restrict__ w2,
              const float* __restrict__ b2, float* __restrict__ out, int n) {
  int wave = threadIdx.x >> 5, lane = threadIdx.x & 31;
  int row = blockIdx.x * 8 + wave;
  if (row >= n) return;
  size_t b = (size_t)row * 64;
  float p = T[b + lane] * w2[lane] + T[b + lane + 32] * w2[lane + 32];
#pragma unroll
  for (int off = 16; off > 0; off >>= 1) p += __shfl_xor(p, off, 32);
  if (lane == 0) out[row] = p + b2[0];
}

// ---------------------------------------------------------------------------
// Host orchestration
// ---------------------------------------------------------------------------
namespace {
constexpr int NU = 100000, NTX = 400000, NE = 500000, NTGT = 50000, H = GNN_H;

inline void gemm128(const float* X, const int* gidx, const float* W, const float* bias,
                    float* Y, int M, int relu, int accum, hipStream_t s) {
  int g = (M + STEPS * 16 - 1) / (STEPS * 16);
  if (gidx)
    gemm_wmma_bk<128, false><<<g, 256, 0, s>>>(X, gidx, W, bias, Y, M, relu, accum);
  else
    gemm_wmma_bk<128, (bool)HAVE_TDM><<<g, 256, 0, s>>>(X, nullptr, W, bias, Y, M, relu, accum);
}
inline void gemm64(const float* X, const int* gidx, const float* W, const float* bias,
                   float* Y, int M, int relu, int accum, hipStream_t s) {
  int g = (M + STEPS * 32 - 1) / (STEPS * 32);
  gemm_wmma_bk<64, false><<<g, 256, 0, s>>>(X, gidx, W, bias, Y, M, relu, accum);
}
inline void zero(float* p, size_t n, hipStream_t s) {
  zero_f32<<<(unsigned)((n + 255) / 256), 256, 0, s>>>(p, n);
}
}  // namespace

extern "C" void kernel_launch(void* const* d_in, const int* in_sizes, int n_in,
                              void* d_out, int out_size, void* d_ws, size_t ws_size,
                              hipStream_t stream) {
  // ---- inputs (setup_inputs() insertion order, params pytree in insertion order)
  const float* emb_user = (const float*)d_in[0];
  const float* emb_txn  = (const float*)d_in[1];
  struct Layer {
    const float *pred_w1, *pred_b1, *pred_w2, *pred_b2, *upd_w, *upd_b;
    const float *prmp_ln_g, *prmp_ln_b;
    const float *sage_rev_wl, *sage_rev_bl, *sage_rev_wr;
    const float *sage_f2_wl, *sage_f2_bl, *sage_f2_wr;
    const float *ln_user_g, *ln_user_b, *ln_txn_g, *ln_txn_b;
  } L[2];
  int p = 2;
  for (int l = 0; l < 2; ++l) {
    L[l].pred_w1     = (const float*)d_in[p++];
    L[l].pred_b1     = (const float*)d_in[p++];
    L[l].pred_w2     = (const float*)d_in[p++];
    L[l].pred_b2     = (const float*)d_in[p++];
    L[l].upd_w       = (const float*)d_in[p++];
    L[l].upd_b       = (const float*)d_in[p++];
    L[l].prmp_ln_g   = (const float*)d_in[p++];
    L[l].prmp_ln_b   = (const float*)d_in[p++];
    L[l].sage_rev_wl = (const float*)d_in[p++];
    L[l].sage_rev_bl = (const float*)d_in[p++];
    L[l].sage_rev_wr = (const float*)d_in[p++];
    L[l].sage_f2_wl  = (const float*)d_in[p++];
    L[l].sage_f2_bl  = (const float*)d_in[p++];
    L[l].sage_f2_wr  = (const float*)d_in[p++];
    L[l].ln_user_g   = (const float*)d_in[p++];
    L[l].ln_user_b   = (const float*)d_in[p++];
    L[l].ln_txn_g    = (const float*)d_in[p++];
    L[l].ln_txn_b    = (const float*)d_in[p++];
  }
  const float* head_w1 = (const float*)d_in[p++];
  const float* head_b1 = (const float*)d_in[p++];
  const float* head_w2 = (const float*)d_in[p++];
  const float* head_b2 = (const float*)d_in[p++];
  const int* fk_src  = (const int*)d_in[p++];
  const int* fk_dst  = (const int*)d_in[p++];
  const int* rev_src = (const int*)d_in[p++];
  const int* rev_dst = (const int*)d_in[p++];
  const int* f2_src  = (const int*)d_in[p++];
  const int* f2_dst  = (const int*)d_in[p++];
  const int* tgt     = (const int*)d_in[p++];

  // ---- workspace arena
  float* w = (float*)d_ws;
  size_t off = 0;
  auto alloc = [&](size_t n) { float* q = w + off; off += n; return q; };
  float* HU[2] = {alloc((size_t)NU * H), alloc((size_t)NU * H)};
  float* HT[2] = {alloc((size_t)NTX * H), alloc((size_t)NTX * H)};
  float* EB0   = alloc((size_t)NE * H);
  float* EB1   = alloc((size_t)NE * H);
  float* SEGU  = alloc((size_t)NU * H);
  float* SEGT  = alloc((size_t)NTX * H);
  float* PRMPU = alloc((size_t)NU * H);
  float* SAGEU = alloc((size_t)NU * H);
  float* SAGET = alloc((size_t)NTX * H);
  float* CNTU  = alloc(NU);
  float* CNTT  = alloc(NTX);
  float* HEADT = alloc((size_t)NTGT * 64);
  (void)ws_size; (void)in_sizes; (void)n_in; (void)out_size;

  (void)hipMemcpyAsync(HU[0], emb_user, (size_t)NU * H * sizeof(float),
                       hipMemcpyDeviceToDevice, stream);
  (void)hipMemcpyAsync(HT[0], emb_txn, (size_t)NTX * H * sizeof(float),
                       hipMemcpyDeviceToDevice, stream);

  const int edgeGrid = ((NE * 32) + 255) / 256;
  int cur = 0;
  for (int l = 0; l < 2; ++l) {
    const Layer& lp = L[l];
    float *hu = HU[cur], *ht = HT[cur];
    float *hun = HU[1 - cur], *htn = HT[1 - cur];

    // --- PRMP on FK edges (gather fused into GEMM A-load) ---
    gemm128(hu, fk_src, lp.pred_w1, lp.pred_b1, EB0, NE, /*relu=*/1, /*accum=*/0, stream);
    gemm128(EB0, nullptr, lp.pred_w2, lp.pred_b2, EB1, NE, 0, 0, stream);
    zero(SEGU, (size_t)NU * H, stream);
    zero(CNTU, NU, stream);
    scatter_residual<<<edgeGrid, 256, 0, stream>>>(EB1, ht, fk_src, fk_dst, SEGU, CNTU, NE);
    seg_div<<<((NU * 32) + 255) / 256, 256, 0, stream>>>(SEGU, CNTU, NU);
    (void)hipMemcpyAsync(PRMPU, hu, (size_t)NU * H * sizeof(float),
                         hipMemcpyDeviceToDevice, stream);
    gemm128(SEGU, nullptr, lp.upd_w, lp.upd_b, PRMPU, NU, 0, /*accum=*/1, stream);
    ln_kernel<<<(NU + 7) / 8, 256, 0, stream>>>(PRMPU, nullptr, 1.0f,
                                                lp.prmp_ln_g, lp.prmp_ln_b, PRMPU, NU, 0);

    // --- SAGE txn -> user ---
    zero(SEGU, (size_t)NU * H, stream);
    zero(CNTU, NU, stream);
    scatter_gather<<<edgeGrid, 256, 0, stream>>>(ht, rev_src, rev_dst, SEGU, CNTU, NE);
    seg_div<<<((NU * 32) + 255) / 256, 256, 0, stream>>>(SEGU, CNTU, NU);
    gemm128(SEGU, nullptr, lp.sage_rev_wl, lp.sage_rev_bl, SAGEU, NU, 0, 0, stream);
    gemm128(hu, nullptr, lp.sage_rev_wr, nullptr, SAGEU, NU, 0, /*accum=*/1, stream);

    // --- SAGE user -> txn ---
    zero(SEGT, (size_t)NTX * H, stream);
    zero(CNTT, NTX, stream);
    scatter_gather<<<edgeGrid, 256, 0, stream>>>(hu, f2_src, f2_dst, SEGT, CNTT, NE);
    seg_div<<<((NTX * 32) + 255) / 256, 256, 0, stream>>>(SEGT, CNTT, NTX);
    gemm128(SEGT, nullptr, lp.sage_f2_wl, lp.sage_f2_bl, SAGET, NTX, 0, 0, stream);
    gemm128(ht, nullptr, lp.sage_f2_wr, nullptr, SAGET, NTX, 0, /*accum=*/1, stream);

    // --- merge: per-type LN((prmp+sage)*0.5) + ReLU ---
    ln_kernel<<<(NU + 7) / 8, 256, 0, stream>>>(PRMPU, SAGEU, 0.5f,
                                                lp.ln_user_g, lp.ln_user_b, hun, NU, 1);
    ln_kernel<<<(NTX + 7) / 8, 256, 0, stream>>>(ht, SAGET, 0.5f,
                                                 lp.ln_txn_g, lp.ln_txn_b, htn, NTX, 1);
    cur = 1 - cur;
  }

  // --- head: out = relu(hu[tgt] @ w1 + b1) @ w2 + b2 ---
  gemm64(HU[cur], tgt, head_w1, head_b1, HEADT, NTGT, /*relu=*/1, 0, stream);
  head_dot<<<(NTGT + 7) / 8, 256, 0, stream>>>(HEADT, head_w2, head_b2, (float*)d_out, NTGT);
}